// SimpleNet_91036126806212
// MI455X (gfx1250) — compile-verified
//
#include <hip/hip_runtime.h>

// ---------------------------------------------------------------------------
// MI455X (gfx1250) implementation.
//   B=8, C=256, L=4096, K=65, PAD=32
//
//  K1: depthwise complex correlation as GEMM vs Toeplitz kernel matrix,
//      V_WMMA_F32_16X16X4_F32, 16(chan) x 16(pos) tiles, Kdim=80.
//      Also emits per-workgroup partial sums for the complex std.
//  K2: per-batch reduction -> inv_std[b].
//  K3: normalize * mag * exp(-i*ang), then 256-pt FFT along C as two
//      16-point DFT stages on the matrix pipe (complex 16x16x16 matmuls,
//      4x v_wmma_f32_16x16x4_f32 per real matmul), twiddles on VALU,
//      fftshift via k ^ 128. y array lives in d_out (K1 writes, K3 rewrites).
// ---------------------------------------------------------------------------

typedef __attribute__((ext_vector_type(2))) float v2f;
typedef __attribute__((ext_vector_type(8))) float v8f;

#define PI_F 3.14159265358979323846f

static __device__ __forceinline__ v8f wmma4(v2f a, v2f b, v8f c) {
  // (neg_a, A, neg_b, B, c_mod, C, reuse_a, reuse_b); f32: A/B negate illegal
  return __builtin_amdgcn_wmma_f32_16x16x4_f32(false, a, false, b, (short)0, c,
                                               false, false);
}

// ============================ Kernel 1: conv =================================
// grid (L/128, C/16, B), block 256 (8 waves). Each wave: one 16x16 output
// tile of yr and yi.  y stored interleaved (re,im) into d_out.
__global__ __launch_bounds__(256) void conv_wmma_kernel(
    const float* __restrict__ x, const float* __restrict__ kr,
    const float* __restrict__ ki, float* __restrict__ yio,
    float* __restrict__ partials) {
  __shared__ float xs[16][193];  // 16 chans x 192 window, pad->193 (no bank cf)
  __shared__ float krs[65];
  __shared__ float kis[65];
  __shared__ float red[8][3];

  const int tid = threadIdx.x;
  const int l0 = blockIdx.x * 128;   // chunk start position
  const int c0 = blockIdx.y * 16;    // channel tile
  const int b  = blockIdx.z;

  if (tid < 65) { krs[tid] = kr[tid]; kis[tid] = ki[tid]; }

  // window covers l in [l0-32, l0+160), zero padded at the edges
  const float* xrow0 = x + ((size_t)(b * 256 + c0)) * 4096;
  __builtin_prefetch(xrow0 + l0, 0, 1);  // -> global_prefetch_b8
  for (int i = tid; i < 16 * 192; i += 256) {
    int c = i / 192, t = i % 192;
    int gl = l0 - 32 + t;
    float v = 0.f;
    if (gl >= 0 && gl < 4096) v = xrow0[(size_t)c * 4096 + gl];
    xs[c][t] = v;
  }
  __syncthreads();

  const int wave = tid >> 5, lane = tid & 31;
  const int half = lane >> 4;      // K-pair select (ISA A/B frag layout)
  const int q    = lane & 15;      // M (A) or N (B)
  const int pos0 = wave * 16;      // this wave's 16 output positions

  v8f accr = {};
  v8f acci = {};
#pragma unroll 4
  for (int base = 0; base < 80; base += 4) {
    const int kk = base + 2 * half;
    v2f a;                                   // A[m=q, kk], A[m=q, kk+1]
    a.x = xs[q][pos0 + kk];
    a.y = xs[q][pos0 + kk + 1];
    const int d0 = kk - q, d1 = d0 + 1;      // Toeplitz: T[kk,n] = k[kk-n]
    v2f br, bi;
    br.x = (d0 >= 0 && d0 < 65) ? krs[d0] : 0.f;
    br.y = (d1 >= 0 && d1 < 65) ? krs[d1] : 0.f;
    bi.x = (d0 >= 0 && d0 < 65) ? kis[d0] : 0.f;
    bi.y = (d1 >= 0 && d1 < 65) ? kis[d1] : 0.f;
    accr = wmma4(a, br, accr);
    acci = wmma4(a, bi, acci);
  }

  // D layout: VGPR r, lanes 0-15 -> M=r, lanes 16-31 -> M=r+8, N = lane&15.
  // y = yr - i*conv_i  (matched-filter conjugate)
  float sr = 0.f, si = 0.f, sq = 0.f;
  const int gl = l0 + pos0 + q;
#pragma unroll
  for (int r = 0; r < 8; ++r) {
    const int m = r + 8 * half;
    const float yr = accr[r];
    const float yi = -acci[r];
    sr += yr;
    si += yi;
    sq += yr * yr + yi * yi;
    float2 v = make_float2(yr, yi);
    *(float2*)&yio[(((size_t)(b * 256 + c0 + m)) * 4096 + gl) * 2] = v;
  }

  // wave32 reduction, then cross-wave via LDS
#pragma unroll
  for (int off = 16; off > 0; off >>= 1) {
    sr += __shfl_xor(sr, off);
    si += __shfl_xor(si, off);
    sq += __shfl_xor(sq, off);
  }
  if (lane == 0) { red[wave][0] = sr; red[wave][1] = si; red[wave][2] = sq; }
  __syncthreads();
  if (tid == 0) {
    float tr = 0.f, ti = 0.f, tq = 0.f;
#pragma unroll
    for (int w = 0; w < 8; ++w) { tr += red[w][0]; ti += red[w][1]; tq += red[w][2]; }
    size_t pid =
        ((size_t)blockIdx.z * gridDim.y + blockIdx.y) * gridDim.x + blockIdx.x;
    partials[pid * 3 + 0] = tr;
    partials[pid * 3 + 1] = ti;
    partials[pid * 3 + 2] = tq;
  }
}

// ============================ Kernel 2: stats ================================
// one block per batch: 512 partial triples -> inv_std[b]
__global__ __launch_bounds__(256) void stats_kernel(
    const float* __restrict__ partials, float* __restrict__ inv_std) {
  __shared__ float s0[256], s1[256], s2[256];
  const int b = blockIdx.x, tid = threadIdx.x;
  const size_t base = (size_t)b * 512 * 3;
  float a0 = partials[base + (size_t)tid * 3 + 0] +
             partials[base + (size_t)(tid + 256) * 3 + 0];
  float a1 = partials[base + (size_t)tid * 3 + 1] +
             partials[base + (size_t)(tid + 256) * 3 + 1];
  float a2 = partials[base + (size_t)tid * 3 + 2] +
             partials[base + (size_t)(tid + 256) * 3 + 2];
  s0[tid] = a0; s1[tid] = a1; s2[tid] = a2;
  __syncthreads();
  for (int off = 128; off > 0; off >>= 1) {
    if (tid < off) {
      s0[tid] += s0[tid + off];
      s1[tid] += s1[tid + off];
      s2[tid] += s2[tid + off];
    }
    __syncthreads();
  }
  if (tid == 0) {
    const float n = 256.f * 4096.f;
    const float mr = s0[0] / n, mi = s1[0] / n;
    const float var = (s2[0] - n * (mr * mr + mi * mi)) / (n - 1.f);
    inv_std[b] = rsqrtf(var);
  }
}

// ============================ Kernel 3: FFT ==================================
// grid (L/8, B), block 256 (8 waves). Each wave = one l column (8 per block):
// 256-pt DFT over C via two WMMA 16x16 complex-matmul stages + twiddles.
__global__ __launch_bounds__(256) void fft_wmma_kernel(
    const float* __restrict__ wmag, const float* __restrict__ wang,
    const float* __restrict__ inv_std, float* __restrict__ yio) {
  __shared__ float yr_s[256 * 9];      // [c][lw], stride 9
  __shared__ float yi_s[256 * 9];
  __shared__ float facr[256], faci[256];
  __shared__ float ub[8][2][16][17];   // per-wave D->A relayout buffer

  const int tid = threadIdx.x;
  const int l0 = blockIdx.x * 8;
  const int b = blockIdx.y;
  const float s = inv_std[b];

  {  // per-channel factor: inv_std * mag * exp(-i*angle)
    const float wm = wmag[tid];
    const float wa = wang[tid];
    facr[tid] = s * wm * __cosf(wa);
    faci[tid] = -s * wm * __sinf(wa);
  }
  __syncthreads();

  for (int i = tid; i < 256 * 8; i += 256) {
    const int c = i >> 3, lw = i & 7;
    float2 v = *(const float2*)&yio[(((size_t)(b * 256 + c)) * 4096 + l0 + lw) * 2];
    const float fr = facr[c], fi = faci[c];
    yr_s[c * 9 + lw] = v.x * fr - v.y * fi;
    yi_s[c * 9 + lw] = v.x * fi + v.y * fr;
  }
  __syncthreads();

  const int wave = tid >> 5, lane = tid & 31;
  const int half = lane >> 4;
  const int q = lane & 15;
  const int lw = wave;

  // DFT-16 matrix F[p,q]=exp(-i*pi*p*q/8); symmetric -> same frags serve the
  // A-layout (stage1) and B-layout (stage2). Fn = -Fi (f32 WMMA has no A-neg).
  v2f Fr[4], Fi[4], Fn[4];
#pragma unroll
  for (int ch = 0; ch < 4; ++ch) {
    const int k0 = 4 * ch + 2 * half;
    const float a0 = -(PI_F / 8.f) * (float)(q * k0);
    const float a1 = -(PI_F / 8.f) * (float)(q * (k0 + 1));
    Fr[ch].x = __cosf(a0); Fr[ch].y = __cosf(a1);
    Fi[ch].x = __sinf(a0); Fi[ch].y = __sinf(a1);
    Fn[ch].x = -Fi[ch].x;  Fn[ch].y = -Fi[ch].y;
  }

  // X[n1][n2] = y[c = 16*n1 + n2], B-layout fragments
  v2f Bxr[4], Bxi[4];
#pragma unroll
  for (int ch = 0; ch < 4; ++ch) {
    const int k0 = 4 * ch + 2 * half;
    const int ca = 16 * k0 + q, cb = 16 * (k0 + 1) + q;
    Bxr[ch].x = yr_s[ca * 9 + lw]; Bxr[ch].y = yr_s[cb * 9 + lw];
    Bxi[ch].x = yi_s[ca * 9 + lw]; Bxi[ch].y = yi_s[cb * 9 + lw];
  }

  // stage 1: T[k1,n2] = sum_n1 F[k1,n1] * X[n1,n2]   (complex)
  v8f Tr = {}, Ti = {};
#pragma unroll
  for (int ch = 0; ch < 4; ++ch) {
    Tr = wmma4(Fr[ch], Bxr[ch], Tr);
    Tr = wmma4(Fn[ch], Bxi[ch], Tr);
    Ti = wmma4(Fr[ch], Bxi[ch], Ti);
    Ti = wmma4(Fi[ch], Bxr[ch], Ti);
  }

  // twiddle W256^{n2*k1}; write D-layout -> ub[m][n] for A-layout reload
#pragma unroll
  for (int r = 0; r < 8; ++r) {
    const int m = r + 8 * half;  // k1
    const int n = q;             // n2
    const float ang = -(2.f * PI_F / 256.f) * (float)(m * n);
    const float cc = __cosf(ang), ss = __sinf(ang);
    const float tr = Tr[r], ti = Ti[r];
    ub[wave][0][m][n] = tr * cc - ti * ss;
    ub[wave][1][m][n] = tr * ss + ti * cc;
  }
  // per-wave buffer: same-wave DS ordering (compiler inserts s_wait_dscnt)
  v2f Ur[4], Ui[4];
#pragma unroll
  for (int ch = 0; ch < 4; ++ch) {
    const int k0 = 4 * ch + 2 * half;
    Ur[ch].x = ub[wave][0][q][k0]; Ur[ch].y = ub[wave][0][q][k0 + 1];
    Ui[ch].x = ub[wave][1][q][k0]; Ui[ch].y = ub[wave][1][q][k0 + 1];
  }

  // stage 2: R[k1,k2] = sum_n2 U[k1,n2] * F[n2,k2]   (F symmetric)
  v8f Rr = {}, Ri = {};
#pragma unroll
  for (int ch = 0; ch < 4; ++ch) {
    Rr = wmma4(Ur[ch], Fr[ch], Rr);
    Rr = wmma4(Ui[ch], Fn[ch], Rr);
    Ri = wmma4(Ur[ch], Fi[ch], Ri);
    Ri = wmma4(Ui[ch], Fr[ch], Ri);
  }

  // X[k1 + 16*k2], fftshift: k -> k ^ 128; only this wave's column touched
#pragma unroll
  for (int r = 0; r < 8; ++r) {
    const int k = ((r + 8 * half) + 16 * q) ^ 128;
    yr_s[k * 9 + lw] = Rr[r];
    yi_s[k * 9 + lw] = Ri[r];
  }
  __syncthreads();

  for (int i = tid; i < 256 * 8; i += 256) {
    const int c = i >> 3, lwi = i & 7;
    float2 v = make_float2(yr_s[c * 9 + lwi], yi_s[c * 9 + lwi]);
    *(float2*)&yio[(((size_t)(b * 256 + c)) * 4096 + l0 + lwi) * 2] = v;
  }
}

// ============================== launcher =====================================
extern "C" void kernel_launch(void* const* d_in, const int* in_sizes, int n_in,
                              void* d_out, int out_size, void* d_ws,
                              size_t ws_size, hipStream_t stream) {
  (void)in_sizes; (void)n_in; (void)out_size; (void)ws_size;
  const float* x  = (const float*)d_in[0];
  const float* kr = (const float*)d_in[1];
  const float* ki = (const float*)d_in[2];
  const float* wm = (const float*)d_in[3];
  const float* wa = (const float*)d_in[4];
  float* yio = (float*)d_out;              // interleaved complex [B,C,L]
  float* partials = (float*)d_ws;          // 4096 * 3 floats
  float* inv_std = partials + 4096 * 3;    // 8 floats

  dim3 g1(32, 16, 8);                      // L/128, C/16, B
  conv_wmma_kernel<<<g1, 256, 0, stream>>>(x, kr, ki, yio, partials);
  stats_kernel<<<8, 256, 0, stream>>>(partials, inv_std);
  dim3 g3(512, 8);                         // L/8, B
  fft_wmma_kernel<<<g3, 256, 0, stream>>>(wm, wa, inv_std, yio);
}